// EdgeUNet_17609365914510
// MI455X (gfx1250) — compile-verified
//
#include <hip/hip_runtime.h>

typedef __attribute__((ext_vector_type(2))) float v2f;
typedef __attribute__((ext_vector_type(8))) float v8f;

#define NV2 163842
#define NV1 40962
#define NV0 10242
#define ALPHA 0.3f

// virtual-concat float4 accessor: channels [0,CA) from fA (optionally gathered
// through mapA), channels [CA, CA+CB) from fB. c4 is a multiple of 4 and CA/CB
// are multiples of 4, so a chunk never straddles the seam.
__device__ __forceinline__ float4 feat_get4(const float* __restrict__ fA,
                                            const int* __restrict__ mapA, int CA,
                                            const float* __restrict__ fB, int CB,
                                            int v, int c4) {
  if (c4 < CA) {
    int u = mapA ? mapA[v] : v;
    return *(const float4*)(fA + (long)u * CA + c4);
  }
  return *(const float4*)(fB + (long)v * CB + (c4 - CA));
}

// Fused EdgeConv, full fp32: gather 6-neighbor mean (float4-vectorized), build
// x=[local | mean-local] tile in LDS (fp32), then V_WMMA_F32_16X16X4_F32 over
// K-tiles with B fragments streamed straight from global (W is L2-hot), bias
// folded into accumulator init, LeakyReLU on store.
// Block = 64 threads (2 waves), one 16-vertex M-tile per block.
__global__ __launch_bounds__(64) void edgeconv_wmma(
    const float* __restrict__ fA, const int* __restrict__ mapA, int CA,
    const float* __restrict__ fB, int CB,
    const int* __restrict__ src,
    const float* __restrict__ W, const float* __restrict__ bias,
    float* __restrict__ out, int nv, int Cin, int Cout)
{
  extern __shared__ __align__(16) char smem[];
  int*   sidx = (int*)smem;                  // [16][6] neighbor indices
  float* xl   = (float*)(smem + 384);        // [16][K2] x-tile
  const int K2   = 2 * Cin;                  // GEMM K dimension
  const int tid  = threadIdx.x;
  const int tile = blockIdx.x;

  // stage neighbor indices (src row per tile vertex)
  for (int i = tid; i < 96; i += 64) {
    int r = i / 6, j = i - r * 6;
    int v = tile * 16 + r; if (v >= nv) v = nv - 1;
    sidx[i] = src[v * 6 + j];
  }
  __syncthreads();

  // build x tile, 4 channels per thread-iteration (b128 loads)
  const int C4 = Cin >> 2;
  for (int i = tid; i < 16 * C4; i += 64) {
    int r = i / C4, c4 = (i - r * C4) << 2;
    int v = tile * 16 + r; if (v >= nv) v = nv - 1;   // clamp tail (store guarded)
    float4 loc = feat_get4(fA, mapA, CA, fB, CB, v, c4);
    float sx = 0.f, sy = 0.f, sz = 0.f, sw = 0.f;
    #pragma unroll
    for (int j = 0; j < 6; ++j) {
      float4 t = feat_get4(fA, mapA, CA, fB, CB, sidx[r * 6 + j], c4);
      sx += t.x; sy += t.y; sz += t.z; sw += t.w;
    }
    const float inv6 = 1.0f / 6.0f;
    float4 d = make_float4(sx * inv6 - loc.x, sy * inv6 - loc.y,
                           sz * inv6 - loc.z, sw * inv6 - loc.w);
    *(float4*)(xl + r * K2 + c4)       = loc;
    *(float4*)(xl + r * K2 + Cin + c4) = d;
  }
  __syncthreads();

  const int lane = tid & 31;
  const int wave = tid >> 5;
  const int g    = lane >> 4;    // half-wave group
  const int nl   = lane & 15;    // m-row (for A) / n-col (for B,D) selector
  const int nK   = K2 >> 2;      // K-tiles of 4
  const int nNT  = Cout >> 4;    // N-tiles of 16

  for (int nt = wave; nt < nNT; nt += 2) {
    const int n = nt * 16 + nl;
    const float bv = bias[n];
    v8f acc;
    #pragma unroll
    for (int j = 0; j < 8; ++j) acc[j] = bv;

    // A fragment (32-bit A 16x4): lane holds {A[m][k0+2g], A[m][k0+2g+1]}, m = nl
    // B fragment (32-bit B 4x16): lane holds {B[k0+2g][n], B[k0+2g+1][n]}
    //   = {W[n][k0+2g], W[n][k0+2g+1]} -> contiguous pair in W row n (global)
    const float* __restrict__ ap = xl + nl * K2 + 2 * g;
    const float* __restrict__ bp = W + (long)n * K2 + 2 * g;
    for (int kt = 0; kt < nK; ++kt) {
      v2f a = *(const v2f*)(ap + 4 * kt);
      v2f b = *(const v2f*)(bp + 4 * kt);
      acc = __builtin_amdgcn_wmma_f32_16x16x4_f32(
          /*neg_a=*/false, a, /*neg_b=*/false, b,
          /*c_mod=*/(short)0, acc, /*reuse_a=*/false, /*reuse_b=*/false);
    }

    // D layout: VGPR j -> (M = j + 8g, N = nl); store with LeakyReLU
    #pragma unroll
    for (int j = 0; j < 8; ++j) {
      int m = tile * 16 + j + 8 * g;
      if (m < nv) {
        float y = acc[j];
        y = (y >= 0.f) ? y : ALPHA * y;
        out[(long)m * Cout + n] = y;
      }
    }
  }
}

// ---- segment_max pooling via order-preserving uint mapping + atomicMax ----
__device__ __forceinline__ unsigned ford(float f) {
  unsigned u = __float_as_uint(f);
  return (u & 0x80000000u) ? ~u : (u | 0x80000000u);
}
__device__ __forceinline__ float funord(unsigned u) {
  unsigned b = (u & 0x80000000u) ? (u & 0x7FFFFFFFu) : ~u;
  return __uint_as_float(b);
}

__global__ void pool_init(unsigned* __restrict__ p, int n) {
  int i = blockIdx.x * blockDim.x + threadIdx.x;
  if (i < n) p[i] = 0x007FFFFFu;   // ford(-inf)
}
__global__ void pool_scatter(const float* __restrict__ f, const int* __restrict__ map,
                             unsigned* __restrict__ buf, int nchild, int C) {
  int i = blockIdx.x * blockDim.x + threadIdx.x;
  if (i >= nchild * C) return;
  int child = i / C, c = i - child * C;
  atomicMax(&buf[(long)map[child] * C + c], ford(f[i]));
}
__global__ void pool_decode(unsigned* __restrict__ buf, int n) {
  int i = blockIdx.x * blockDim.x + threadIdx.x;
  if (i < n) {
    float v = funord(buf[i]);
    ((float*)buf)[i] = v;          // decode in place; buffer becomes float
  }
}

extern "C" void kernel_launch(void* const* d_in, const int* in_sizes, int n_in,
                              void* d_out, int out_size, void* d_ws, size_t ws_size,
                              hipStream_t stream) {
  (void)in_sizes; (void)n_in; (void)out_size; (void)ws_size;
  const float* features = (const float*)d_in[0];
  const int*   src2     = (const int*)d_in[1];
  const int*   src1     = (const int*)d_in[3];
  const int*   src0     = (const int*)d_in[5];
  const int*   p21      = (const int*)d_in[7];
  const int*   p10      = (const int*)d_in[8];
  const float* w_enc0 = (const float*)d_in[9],  *b_enc0 = (const float*)d_in[10];
  const float* w_enc1 = (const float*)d_in[11], *b_enc1 = (const float*)d_in[12];
  const float* w_enc2 = (const float*)d_in[13], *b_enc2 = (const float*)d_in[14];
  const float* w_dec0 = (const float*)d_in[15], *b_dec0 = (const float*)d_in[16];
  const float* w_dec1 = (const float*)d_in[17], *b_dec1 = (const float*)d_in[18];

  // workspace arena (all offsets 16B-aligned)
  char* ws = (char*)d_ws;
  size_t off = 0;
  float*    f2 = (float*)(ws + off); off += (size_t)NV2 * 32 * 4;   // 20.97 MB
  float*    f1 = (float*)(ws + off); off += (size_t)NV1 * 64 * 4;   // 10.49 MB
  float*    f0 = (float*)(ws + off); off += (size_t)NV0 * 128 * 4;  //  5.24 MB
  float*    d1 = (float*)(ws + off); off += (size_t)NV1 * 64 * 4;   // 10.49 MB
  unsigned* P  = (unsigned*)(ws + off);                             //  5.24 MB (pool scratch, reused)

  auto ec = [&](const float* fA, const int* mapA, int CA, const float* fB, int CB,
                const int* src, const float* W, const float* b, float* out,
                int nv, int Cin, int Cout) {
    int K2 = 2 * Cin;
    size_t shmem = 384 + (size_t)(16 * K2) * sizeof(float);
    int grid = (nv + 15) / 16;
    edgeconv_wmma<<<grid, 64, shmem, stream>>>(fA, mapA, CA, fB, CB, src, W, b,
                                               out, nv, Cin, Cout);
  };
  auto pool = [&](const float* fchild, const int* map, int nchild, int nparent, int C) {
    int nc = nchild * C, np = nparent * C;
    pool_init   <<<(np + 255) / 256, 256, 0, stream>>>(P, np);
    pool_scatter<<<(nc + 255) / 256, 256, 0, stream>>>(fchild, map, P, nchild, C);
    pool_decode <<<(np + 255) / 256, 256, 0, stream>>>(P, np);
  };

  // encoder
  ec(features, nullptr, 16, nullptr, 0, src2, w_enc0, b_enc0, f2, NV2, 16, 32);
  pool(f2, p21, NV2, NV1, 32);                                   // P = f1_pooled [NV1,32]
  ec((const float*)P, nullptr, 32, nullptr, 0, src1, w_enc1, b_enc1, f1, NV1, 32, 64);
  pool(f1, p10, NV1, NV0, 64);                                   // P = f0_pooled [NV0,64]
  ec((const float*)P, nullptr, 64, nullptr, 0, src0, w_enc2, b_enc2, f0, NV0, 64, 128);

  // decoder (skip-concat virtualized in the gather: never materialize u1/u2)
  ec(f0, p10, 128, f1, 64, src1, w_dec0, b_dec0, d1, NV1, 192, 64);          // u1 = [f0[p10] | f1]
  ec(d1, p21, 64, f2, 32, src2, w_dec1, b_dec1, (float*)d_out, NV2, 96, 32); // u2 = [d1[p21] | f2]
}